// MyModel_76115410419996
// MI455X (gfx1250) — compile-verified
//
#include <hip/hip_runtime.h>
#include <hip/hip_bf16.h>

typedef __attribute__((ext_vector_type(2))) float v2f;
typedef __attribute__((ext_vector_type(8))) float v8f;

#define TSTEPS 2048
#define ROWS_PER_WAVE 16
#define WAVES_PER_BLOCK 8
#define ROWS_PER_BLOCK (ROWS_PER_WAVE * WAVES_PER_BLOCK)

__device__ __forceinline__ float fast_rcp(float x) {
    return __builtin_amdgcn_rcpf(x);
}

// SWAPX16: group-of-32 swizzle, xor=0x10, or=0x00, and=0x1f -> offset 0x401F
__device__ __forceinline__ float swz16(float x) {
    return __int_as_float(__builtin_amdgcn_ds_swizzle(__float_as_int(x), 0x401F));
}

__device__ __forceinline__ float sigmoid_f(float x) {
    return fast_rcp(1.0f + __expf(-x));
}

__device__ __forceinline__ float softsign_beta(float x, float beta) {
    return x * fast_rcp(1.0f + fabsf(x)) * beta;
}

__global__ __launch_bounds__(256) void lstm_wmma_kernel(
    const float* __restrict__ x,          // [B, T, 1]
    const float* __restrict__ kernel,     // [1, 16]
    const float* __restrict__ rec_kernel, // [4, 16]
    const float* __restrict__ bias,       // [16]
    const float* __restrict__ dense_w,    // [4, 32]
    const float* __restrict__ dense_b,    // [32]
    const float* __restrict__ beta_p,     // [1]
    float* __restrict__ out,              // [B, 32]
    int B)
{
    const int lane = threadIdx.x & 31;
    const int wave = threadIdx.x >> 5;
    const bool hi  = lane >= 16;
    const int  m   = lane & 15;          // batch column within tile / A-matrix row
    const int  row = blockIdx.x * ROWS_PER_BLOCK + wave * ROWS_PER_WAVE + m;
    const int  rowc = (row < B) ? row : 0;

    const float beta = beta_p[0];
    const int kb = hi ? 2 : 0;           // K pair this lane supplies for A
    const int gb = hi ? 8 : 0;           // gate base this lane's D rows cover

    // A = rec_kernel^T (16x4 f32): lane m<16 -> A[m][0..1], lane 16+m -> A[m][2..3]
    // A[m][k] = rec_kernel[k*16 + m]
    v2f amat;
    amat.x = rec_kernel[(kb + 0) * 16 + m];
    amat.y = rec_kernel[(kb + 1) * 16 + m];

    // Per-lane input-kernel and bias slices for the C matrix (z^T layout:
    // VGPR v of this lane corresponds to gate gb+v).
    float kv[8], bv[8];
#pragma unroll
    for (int v = 0; v < 8; ++v) {
        kv[v] = kernel[gb + v];
        bv[v] = bias[gb + v];
    }

    // Split state ownership: low lanes keep (c0,c1,h0,h1), high lanes (c2,c3,h2,h3).
    // This is exactly what each half must feed into the next step's B-matrix.
    float h[2] = {0.f, 0.f};
    float c[2] = {0.f, 0.f};

    const float* xrow = x + (size_t)rowc * TSTEPS;

    for (int t0 = 0; t0 < TSTEPS; t0 += 4) {
        if ((t0 & 31) == 0) {
            // Speculative prefetch ~5 cachelines ahead (OOB prefetch is dropped).
            __builtin_prefetch(xrow + t0 + 160, 0, 1);
        }
        const float4 xq = *(const float4*)(xrow + t0);
        const float xs[4] = {xq.x, xq.y, xq.z, xq.w};
#pragma unroll
        for (int u = 0; u < 4; ++u) {
            const float xt = xs[u];

            // C = x_t * kernel + bias  (z^T layout)
            v8f cm;
#pragma unroll
            for (int v = 0; v < 8; ++v) cm[v] = fmaf(xt, kv[v], bv[v]);

            // B = h^T (4x16): lane n<16 supplies h0,h1; lane 16+n supplies h2,h3
            // (each half owns exactly those states -> no selects)
            v2f bm;
            bm.x = h[0];
            bm.y = h[1];

            // z^T = A x B + C   -> v_wmma_f32_16x16x4_f32
            v8f d = __builtin_amdgcn_wmma_f32_16x16x4_f32(
                false, amat, false, bm, (short)0, cm, false, false);

            // Activations on own gates: low lanes have i(0-3),f(4-7);
            // high lanes have g(0-3),o(4-7). g uses softsign*beta, rest sigmoid.
            float s[8];
#pragma unroll
            for (int v = 0; v < 4; ++v) {
                const float z  = d[v];
                const float sg = sigmoid_f(z);
                const float ss = softsign_beta(z, beta);
                s[v] = hi ? ss : sg;
            }
#pragma unroll
            for (int v = 4; v < 8; ++v) s[v] = sigmoid_f(d[v]);

            // Packed half-exchange: 4 swizzles carry different payloads each way.
            //   low -> high : si_2, si_3, sf_2, sf_3   (s[2],s[3],s[6],s[7])
            //   high -> low : ag_0, ag_1, so_0, so_1   (s[0],s[1],s[4],s[5])
            float u0 = hi ? s[0] : s[2];
            float u1 = hi ? s[1] : s[3];
            float u2 = hi ? s[4] : s[6];
            float u3 = hi ? s[5] : s[7];
            float w0 = swz16(u0);   // low: ag_0 | high: si_2
            float w1 = swz16(u1);   // low: ag_1 | high: si_3
            float w2 = swz16(u2);   // low: so_0 | high: sf_2
            float w3 = swz16(u3);   // low: so_1 | high: sf_3
            const float w[4] = {w0, w1, w2, w3};

            // Each half updates only its two owned states.
#pragma unroll
            for (int p = 0; p < 2; ++p) {
                const float si = hi ? w[p]     : s[p];
                const float sf = hi ? w[2 + p] : s[4 + p];
                const float ag = hi ? s[2 + p] : w[p];
                const float so = hi ? s[6 + p] : w[2 + p];
                const float cn = fmaf(sf, c[p], si * ag);
                c[p] = cn;
                h[p] = so * softsign_beta(cn, beta);
            }
        }
    }

    // Reconstruct full h[4] per lane (2 swizzles), then dense + activation;
    // low lanes write out columns 0-15, high lanes 16-31.
    const float g0 = swz16(h[0]);   // low receives h2 | high receives h0
    const float g1 = swz16(h[1]);   // low receives h3 | high receives h1
    float hf[4];
    hf[0] = hi ? g0 : h[0];
    hf[1] = hi ? g1 : h[1];
    hf[2] = hi ? h[0] : g0;
    hf[3] = hi ? h[1] : g1;

    if (row < B) {
        float* orow = out + (size_t)row * 32;
        const int dbase = hi ? 16 : 0;
#pragma unroll
        for (int dd = 0; dd < 16; ++dd) {
            const int dcol = dbase + dd;
            float vv = dense_b[dcol];
#pragma unroll
            for (int j = 0; j < 4; ++j)
                vv = fmaf(hf[j], dense_w[j * 32 + dcol], vv);
            orow[dcol] = softsign_beta(vv, beta);
        }
    }
}

extern "C" void kernel_launch(void* const* d_in, const int* in_sizes, int n_in,
                              void* d_out, int out_size, void* d_ws, size_t ws_size,
                              hipStream_t stream) {
    const float* x          = (const float*)d_in[0];
    const float* kernel     = (const float*)d_in[1];
    const float* rec_kernel = (const float*)d_in[2];
    const float* bias       = (const float*)d_in[3];
    const float* dense_w    = (const float*)d_in[4];
    const float* dense_b    = (const float*)d_in[5];
    const float* beta       = (const float*)d_in[6];
    float* out              = (float*)d_out;

    const int B = in_sizes[0] / TSTEPS;   // F = 1
    const int blocks = (B + ROWS_PER_BLOCK - 1) / ROWS_PER_BLOCK;

    lstm_wmma_kernel<<<blocks, 256, 0, stream>>>(
        x, kernel, rec_kernel, bias, dense_w, dense_b, beta, out, B);
}